// Psaa_Module_53094385713345
// MI455X (gfx1250) — compile-verified
//
#include <hip/hip_runtime.h>
#include <math.h>

// Problem constants (from reference): N=4, C=512, H=W=64, S=6
#define Nn   4
#define Cc   512
#define Hh   64
#define Wwid 64
#define Ss   6
#define HWp  (Hh * Wwid)          // 4096 pixels per image
#define Kproj (Ss * Cc)           // 3072
#define BN_EPS 1e-5f

typedef __attribute__((ext_vector_type(2))) float v2f;
typedef __attribute__((ext_vector_type(4))) float v4f;
typedef __attribute__((ext_vector_type(8))) float v8f;

// ---------------------------------------------------------------------------
// Kernel A: fused  project -> softmax -> out1 -> energy2 -> softmax -> out2
//           -> out = gamma*out2 + out1, written as [n][c][p] (coalesced).
// One thread per pixel. No LDS; out1 is recomputed in pass 2 from the same
// yv values (pass-2 stack reads hit L2: per-block working set ~3MB << 192MB).
// ---------------------------------------------------------------------------
__global__ __launch_bounds__(256) void psaa_attn_kernel(
    const float* __restrict__ cat,     // [N][S*C][H][W]
    const float* __restrict__ stack,   // [N][C][H][W][S]
    const float* __restrict__ w_proj,  // [S][S*C]
    const float* __restrict__ b_proj,  // [S]
    const float* __restrict__ gamma_p, // [1]
    float* __restrict__ out_ws)        // [N][C][HW]
{
    const int idx = blockIdx.x * blockDim.x + threadIdx.x;   // exact N*HW grid
    const int n = idx / HWp;
    const int p = idx % HWp;

    // ---- energy[s] = sum_k cat[n,k,p] * w_proj[s,k] + b_proj[s] ----
    float e[Ss];
    #pragma unroll
    for (int s = 0; s < Ss; ++s) e[s] = b_proj[s];

    const float* catp = cat + (size_t)n * Kproj * HWp + p;
    for (int k = 0; k < Kproj; ++k) {
        const float v = catp[(size_t)k * HWp];               // coalesced across lanes
        #pragma unroll
        for (int s = 0; s < Ss; ++s)                         // w_proj: wave-uniform (scalar loads)
            e[s] = fmaf(v, w_proj[s * Kproj + k], e[s]);
    }

    // ---- softmax over s -> a1 ----
    float a1[Ss];
    {
        float m = e[0];
        #pragma unroll
        for (int s = 1; s < Ss; ++s) m = fmaxf(m, e[s]);
        float sum = 0.f;
        #pragma unroll
        for (int s = 0; s < Ss; ++s) { a1[s] = expf(e[s] - m); sum += a1[s]; }
        const float r = 1.f / sum;
        #pragma unroll
        for (int s = 0; s < Ss; ++s) a1[s] *= r;
    }

    // ---- pass 1 over yv: energy2[s] = sum_c yv[c,s] * (sum_t yv[c,t]*a1[t]) ----
    const float* stp = stack + ((size_t)n * Cc * HWp + p) * Ss;   // 6 contiguous floats per c
    float e2[Ss];
    #pragma unroll
    for (int s = 0; s < Ss; ++s) e2[s] = 0.f;

    for (int c = 0; c < Cc; ++c) {
        const float* yp = stp + (size_t)c * (HWp * Ss);
        v2f y01 = *(const v2f*)(yp + 0);     // base is 24B multiple -> 8B aligned
        v2f y23 = *(const v2f*)(yp + 2);
        v2f y45 = *(const v2f*)(yp + 4);
        float y[Ss] = { y01.x, y01.y, y23.x, y23.y, y45.x, y45.y };
        float o1 = 0.f;
        #pragma unroll
        for (int s = 0; s < Ss; ++s) o1 = fmaf(y[s], a1[s], o1);
        #pragma unroll
        for (int s = 0; s < Ss; ++s) e2[s] = fmaf(y[s], o1, e2[s]);
    }

    // ---- softmax over s -> a2 ----
    float a2[Ss];
    {
        float m = e2[0];
        #pragma unroll
        for (int s = 1; s < Ss; ++s) m = fmaxf(m, e2[s]);
        float sum = 0.f;
        #pragma unroll
        for (int s = 0; s < Ss; ++s) { a2[s] = expf(e2[s] - m); sum += a2[s]; }
        const float r = 1.f / sum;
        #pragma unroll
        for (int s = 0; s < Ss; ++s) a2[s] *= r;
    }

    // ---- pass 2 over yv (L2 hits): out = gamma*out2 + out1, layout [n][c][p] ----
    const float gamma = gamma_p[0];
    float* dst = out_ws + (size_t)n * Cc * HWp + p;
    for (int c = 0; c < Cc; ++c) {
        const float* yp = stp + (size_t)c * (HWp * Ss);
        v2f y01 = *(const v2f*)(yp + 0);
        v2f y23 = *(const v2f*)(yp + 2);
        v2f y45 = *(const v2f*)(yp + 4);
        float y[Ss] = { y01.x, y01.y, y23.x, y23.y, y45.x, y45.y };
        float o1 = 0.f, o2 = 0.f;
        #pragma unroll
        for (int s = 0; s < Ss; ++s) {
            o1 = fmaf(y[s], a1[s], o1);
            o2 = fmaf(y[s], a2[s], o2);
        }
        dst[(size_t)c * HWp] = fmaf(gamma, o2, o1);          // coalesced store
    }
}

// ---------------------------------------------------------------------------
// Kernel B: fused 1x1 conv (C->C GEMM) + BatchNorm(eval) + ReLU using
// V_WMMA_F32_16X16X4_F32 with 4x4 register blocking.
// One wave32 computes a 64(pixel) x 64(out-ch) macro-tile as 16 WMMA subtiles
// (16 x v8f accumulators), K-loop over C=512 in steps of 4:
//   A[i] frag: out_ws[n, kb+k, pbase+16i+lane%16]   (coalesced per half-wave)
//   B[j] frag: w_fuse[obase+16j+lane%16, kb+k]      (single b64: k,k+1 contig)
// -> 16 loaded floats per output (vs 64 unblocked); all 16 WMMAs per K-step
// are independent (distinct accumulators) for maximal XDL pipelining.
// The 8 waves of a block share one A macro-tile (same pbase, all 8 o-blocks),
// so the 128KB A tile is WGP$-resident and w_fuse is read once per block.
// D layout (ISA 7.12.2): VGPR v = row M (lanes 0-15: M=v, 16-31: M=8+v),
// lane%16 = column N -> each lane stores 8 contiguous pixels of one channel;
// lanes L and L+16 complete the same 64B line.
// ---------------------------------------------------------------------------
__global__ __launch_bounds__(256) void psaa_fuse_gemm_kernel(
    const float* __restrict__ out_ws,  // [N][C][HW]
    const float* __restrict__ w_fuse,  // [C][C]  (o-major)
    const float* __restrict__ bn_w,
    const float* __restrict__ bn_b,
    const float* __restrict__ bn_m,
    const float* __restrict__ bn_v,
    float* __restrict__ out)           // [N][C][HW]
{
    const int tid  = blockIdx.x * blockDim.x + threadIdx.x;
    const int lane = threadIdx.x & 31;
    const int wave = tid >> 5;                 // 2048 waves total

    const int oblk = wave & 7;                 // 8 macro-tiles of 64 out-channels
    const int pblk = (wave >> 3) & 63;         // 64 macro-tiles of 64 pixels
    const int n    = wave >> 9;                // 4 images

    const int l16  = lane & 15;
    const int half = lane >> 4;                // selects K pair {0,1} vs {2,3}
    const int pbase = pblk * 64;
    const int obase = oblk * 64;
    const int k0 = half * 2;

    // Hoisted per-fragment base pointers
    const float* Ap[4];
    const float* Bp[4];
    #pragma unroll
    for (int i = 0; i < 4; ++i)
        Ap[i] = out_ws + (size_t)n * Cc * HWp + (pbase + i * 16 + l16);
    #pragma unroll
    for (int j = 0; j < 4; ++j)
        Bp[j] = w_fuse + (size_t)(obase + j * 16 + l16) * Cc;

    v8f acc[4][4] = {};
    for (int kb = 0; kb < Cc; kb += 4) {
        v2f a[4], b[4];
        #pragma unroll
        for (int i = 0; i < 4; ++i) {
            a[i].x = Ap[i][(size_t)(kb + k0)     * HWp];
            a[i].y = Ap[i][(size_t)(kb + k0 + 1) * HWp];
        }
        #pragma unroll
        for (int j = 0; j < 4; ++j)
            b[j] = *(const v2f*)(Bp[j] + kb + k0);   // 8B-aligned (kb%4==0, k0 even)

        #pragma unroll
        for (int i = 0; i < 4; ++i)
            #pragma unroll
            for (int j = 0; j < 4; ++j)
                // 8 args: (neg_a, A, neg_b, B, c_mod, C, reuse_a, reuse_b)
                acc[i][j] = __builtin_amdgcn_wmma_f32_16x16x4_f32(
                                false, a[i], false, b[j], (short)0, acc[i][j],
                                false, false);
    }

    // BatchNorm (eval) + ReLU, per output channel (per lane, per j-subtile)
    #pragma unroll
    for (int j = 0; j < 4; ++j) {
        const int o = obase + j * 16 + l16;
        const float inv  = bn_w[o] * rsqrtf(bn_v[o] + BN_EPS);
        const float beta = fmaf(-bn_m[o], inv, bn_b[o]);
        #pragma unroll
        for (int i = 0; i < 4; ++i) {
            float* dst = out + ((size_t)(n * Cc + o)) * HWp
                             + (pbase + i * 16 + half * 8);
            #pragma unroll
            for (int v = 0; v < 8; ++v) {
                const float r = fmaf(acc[i][j][v], inv, beta);
                dst[v] = r > 0.f ? r : 0.f;          // 8 contiguous floats per lane
            }
        }
    }
}

// ---------------------------------------------------------------------------
extern "C" void kernel_launch(void* const* d_in, const int* in_sizes, int n_in,
                              void* d_out, int out_size, void* d_ws, size_t ws_size,
                              hipStream_t stream) {
    const float* cat    = (const float*)d_in[0];
    const float* stack  = (const float*)d_in[1];
    const float* w_proj = (const float*)d_in[2];
    const float* b_proj = (const float*)d_in[3];
    const float* w_fuse = (const float*)d_in[4];
    const float* bn_w   = (const float*)d_in[5];
    const float* bn_b   = (const float*)d_in[6];
    const float* bn_m   = (const float*)d_in[7];
    const float* bn_v   = (const float*)d_in[8];
    const float* gamma  = (const float*)d_in[9];
    float* out = (float*)d_out;

    float* out_ws = (float*)d_ws;              // [N][C][HW] fp32 = 32 MB

    // Kernel A: one thread per pixel, exact grid N*HW = 16384
    psaa_attn_kernel<<<(Nn * HWp) / 256, 256, 0, stream>>>(
        cat, stack, w_proj, b_proj, gamma, out_ws);

    // Kernel B: one wave per 64x64 macro-tile:
    //   4 n * 64 p-blocks * 8 o-blocks = 2048 waves, 8 waves/block -> 256 blocks
    psaa_fuse_gemm_kernel<<<(Nn * (HWp / 64) * (Cc / 64)) / 8, 256, 0, stream>>>(
        out_ws, w_fuse, bn_w, bn_b, bn_m, bn_v, out);
}